// HAN_GNN_20409684590874
// MI455X (gfx1250) — compile-verified
//
#include <hip/hip_runtime.h>
#include <math.h>

typedef float v2f __attribute__((ext_vector_type(2)));
typedef float v8f __attribute__((ext_vector_type(8)));

#define HID 128
#define HEADS 4
#define SEM 512
#define IN_DIM 768
#define OUT_DIM 256
#define N_USER 50000
#define N_LLM 5000
#define NEDGE 400000

__device__ __forceinline__ unsigned lds_off(const void* p) {
  return (unsigned)(uintptr_t)(const __attribute__((address_space(3))) void*)p;
}

// ---------------- WMMA f32 GEMM: C[M,N] = A[M,K] @ B[K,N] (+ bias[N]) ----------------
// block = 256 threads = 8 waves (2 M x 4 N); wave tile 32x32 (4 accumulators);
// block tile 64x128; K staged 32 deep in LDS. B tile staged with async global->LDS copies.
__global__ void gemm_f32_wmma(const float* __restrict__ A, const float* __restrict__ B,
                              float* __restrict__ C, const float* __restrict__ bias,
                              int M, int N, int K) {
  __shared__ float As[64][33];    // +1 pad: column reads hit distinct banks
  __shared__ float Bs[32][132];   // +4 pad (keeps 16B alignment for b128 async stores)
  const int tid  = threadIdx.x;
  const int lane = tid & 31;
  const int wave = tid >> 5;
  const int wm2 = wave >> 2;      // 0..1 -> M offset 32*wm2
  const int wn2 = wave & 3;       // 0..3 -> N offset 32*wn2
  const int blockM = blockIdx.y * 64;
  const int blockN = blockIdx.x * 128;
  const int lm = lane & 15;
  const int lg = lane >> 4;

  v8f acc00 = {0.f,0.f,0.f,0.f,0.f,0.f,0.f,0.f};
  v8f acc01 = {0.f,0.f,0.f,0.f,0.f,0.f,0.f,0.f};
  v8f acc10 = {0.f,0.f,0.f,0.f,0.f,0.f,0.f,0.f};
  v8f acc11 = {0.f,0.f,0.f,0.f,0.f,0.f,0.f,0.f};

  const int arow = tid >> 3;         // 0..31 (rows arow, arow+32)
  const int akq  = (tid & 7) * 4;    // 0..28
  const int brow = tid >> 5;         // 0..7  (rows brow + {0,8,16,24})
  const int bcol = (tid & 31) * 4;   // 0..124

  for (int k0 = 0; k0 < K; k0 += 32) {
    // A tile: synchronous, zero-filled past M (needed for edge blocks)
    #pragma unroll
    for (int rr = 0; rr < 64; rr += 32) {
      int gr = blockM + arow + rr;
      float4 v = make_float4(0.f, 0.f, 0.f, 0.f);
      if (gr < M) v = *(const float4*)(A + (size_t)gr * K + k0 + akq);
      As[arow + rr][akq + 0] = v.x; As[arow + rr][akq + 1] = v.y;
      As[arow + rr][akq + 2] = v.z; As[arow + rr][akq + 3] = v.w;
    }
    // B tile: async global -> LDS (no guards needed: N % 128 == 0, K % 32 == 0)
    #pragma unroll
    for (int rr = 0; rr < 32; rr += 8) {
      const float* gp = B + (size_t)(k0 + brow + rr) * N + blockN + bcol;
      unsigned lo = lds_off(&Bs[brow + rr][bcol]);
      asm volatile("global_load_async_to_lds_b128 %0, %1, off"
                   :: "v"(lo), "v"(gp) : "memory");
    }
    asm volatile("s_wait_asynccnt 0" ::: "memory");
    __syncthreads();

    #pragma unroll
    for (int kk = 0; kk < 32; kk += 4) {
      v2f a0, a1, b0, b1;
      // A 16x4 frag: lane holds (M=lane%16, K = 2*(lane/16) + {0,1})
      a0.x = As[wm2 * 32 + lm][kk + 2 * lg + 0];
      a0.y = As[wm2 * 32 + lm][kk + 2 * lg + 1];
      a1.x = As[wm2 * 32 + 16 + lm][kk + 2 * lg + 0];
      a1.y = As[wm2 * 32 + 16 + lm][kk + 2 * lg + 1];
      // B 4x16 frag: lane holds (N=lane%16, K = 2*(lane/16) + {0,1})
      b0.x = Bs[kk + 2 * lg + 0][wn2 * 32 + lm];
      b0.y = Bs[kk + 2 * lg + 1][wn2 * 32 + lm];
      b1.x = Bs[kk + 2 * lg + 0][wn2 * 32 + 16 + lm];
      b1.y = Bs[kk + 2 * lg + 1][wn2 * 32 + 16 + lm];
      acc00 = __builtin_amdgcn_wmma_f32_16x16x4_f32(false, a0, false, b0, (short)0, acc00, false, false);
      acc01 = __builtin_amdgcn_wmma_f32_16x16x4_f32(false, a0, false, b1, (short)0, acc01, false, false);
      acc10 = __builtin_amdgcn_wmma_f32_16x16x4_f32(false, a1, false, b0, (short)0, acc10, false, false);
      acc11 = __builtin_amdgcn_wmma_f32_16x16x4_f32(false, a1, false, b1, (short)0, acc11, false, false);
    }
    __syncthreads();
  }

  // C/D layout: VGPR r -> lanes 0-15: M=r, lanes 16-31: M=r+8; N = lane%16
  const int col0 = blockN + wn2 * 32 + lm;
  const int col1 = col0 + 16;
  const int rowA = blockM + wm2 * 32 + lg * 8;
  const int rowB = rowA + 16;
  const float bv0 = bias ? bias[col0] : 0.f;
  const float bv1 = bias ? bias[col1] : 0.f;
  #pragma unroll
  for (int r = 0; r < 8; ++r) {
    if (rowA + r < M) {
      C[(size_t)(rowA + r) * N + col0] = acc00[r] + bv0;
      C[(size_t)(rowA + r) * N + col1] = acc01[r] + bv1;
    }
    if (rowB + r < M) {
      C[(size_t)(rowB + r) * N + col0] = acc10[r] + bv0;
      C[(size_t)(rowB + r) * N + col1] = acc11[r] + bv1;
    }
  }
}

// ---------------- a[n*4+h] = sum_c mat[n*512 + h*128 + c] * att[h*128 + c] ----------------
__global__ void rowdot_att(const float* __restrict__ mat, const float* __restrict__ att,
                           float* __restrict__ out, int rows) {
  int gw = (int)((blockIdx.x * blockDim.x + threadIdx.x) >> 5);
  int lane = threadIdx.x & 31;
  if (gw >= rows * HEADS) return;
  int n = gw >> 2, h = gw & 3;
  const float* mp = mat + (size_t)n * SEM + h * HID;
  const float* ap = att + h * HID;
  float p = 0.f;
  for (int c = lane; c < HID; c += 32) p += mp[c] * ap[c];
  for (int off = 16; off > 0; off >>= 1) p += __shfl_xor(p, off, 32);
  if (lane == 0) out[gw] = p;
}

// ---------------- ad[n*4+h] = sum_k x[n*768+k] * vf[k*4+h] ----------------
__global__ void matvec_fold(const float* __restrict__ x, const float* __restrict__ vf,
                            float* __restrict__ out, int rows) {
  int gw = (int)((blockIdx.x * blockDim.x + threadIdx.x) >> 5);
  int lane = threadIdx.x & 31;
  if (gw >= rows * HEADS) return;
  int n = gw >> 2, h = gw & 3;
  const float* xp = x + (size_t)n * IN_DIM;
  float p = 0.f;
  for (int k = lane; k < IN_DIM; k += 32) p += xp[k] * vf[k * 4 + h];
  for (int off = 16; off > 0; off >>= 1) p += __shfl_xor(p, off, 32);
  if (lane == 0) out[gw] = p;
}

// ordered-uint encoding so unsigned atomicMax == float max
__device__ __forceinline__ unsigned fenc(float f) {
  unsigned b = __float_as_uint(f);
  return (b & 0x80000000u) ? ~b : (b | 0x80000000u);
}
__device__ __forceinline__ float fdec(unsigned u) {
  unsigned b = (u & 0x80000000u) ? (u & 0x7FFFFFFFu) : ~u;
  return __uint_as_float(b);
}

__global__ void edge_max(const int* __restrict__ src, const int* __restrict__ dst,
                         const float* __restrict__ as, const float* __restrict__ ad,
                         unsigned* __restrict__ menc) {
  int e = blockIdx.x * 256 + threadIdx.x;
  if (e >= NEDGE) return;
  int s = src[e], d = dst[e];
  #pragma unroll
  for (int h = 0; h < 4; ++h) {
    float v = as[s * 4 + h] + ad[d * 4 + h];
    v = v > 0.f ? v : 0.2f * v;
    atomicMax(&menc[d * 4 + h], fenc(v));
  }
}

__global__ void edge_exp(const int* __restrict__ src, const int* __restrict__ dst,
                         const float* __restrict__ as, const float* __restrict__ ad,
                         const unsigned* __restrict__ menc,
                         float* __restrict__ ex, float* __restrict__ den) {
  int e = blockIdx.x * 256 + threadIdx.x;
  if (e >= NEDGE) return;
  int s = src[e], d = dst[e];
  #pragma unroll
  for (int h = 0; h < 4; ++h) {
    float v = as[s * 4 + h] + ad[d * 4 + h];
    v = v > 0.f ? v : 0.2f * v;
    float m = fdec(menc[d * 4 + h]);
    float ev = expf(v - m);
    ex[e * 4 + h] = ev;
    atomicAdd(&den[d * 4 + h], ev);
  }
}

// 128 threads per edge: thread t handles 4 floats of head h = t/32
__global__ void edge_msg(const int* __restrict__ src, const int* __restrict__ dst,
                         const float* __restrict__ hs, const float* __restrict__ ex,
                         const float* __restrict__ den, float* __restrict__ z) {
  int e = blockIdx.x * 2 + (threadIdx.x >> 7);
  if (e >= NEDGE) return;
  int t = threadIdx.x & 127;
  int s = src[e], d = dst[e];
  int h = t >> 5;
  float alpha = ex[e * 4 + h] / (den[d * 4 + h] + 1e-16f);
  int off = h * HID + (t & 31) * 4;
  float4 hv = *(const float4*)(hs + (size_t)s * SEM + off);
  float* zp = z + (size_t)d * SEM + off;
  atomicAdd(zp + 0, hv.x * alpha);
  atomicAdd(zp + 1, hv.y * alpha);
  atomicAdd(zp + 2, hv.z * alpha);
  atomicAdd(zp + 3, hv.w * alpha);
}

__global__ void bias_elu(float* __restrict__ z, const float* __restrict__ bias, int n) {
  int i = blockIdx.x * 256 + threadIdx.x;
  if (i >= n) return;
  float v = z[i] + bias[i & (SEM - 1)];
  z[i] = v > 0.f ? v : expm1f(v);
}

// w[n*2+p] = tanh(z_p[n] @ W1 + b1) @ W2 ; block = 128 threads, grid = (N_USER, 2)
__global__ void sem_score(const float* __restrict__ z0, const float* __restrict__ z1,
                          const float* __restrict__ W1, const float* __restrict__ b1,
                          const float* __restrict__ W2, float* __restrict__ w) {
  __shared__ float zs[SEM];
  __shared__ float red[128];
  int n = blockIdx.x, p = blockIdx.y;
  const float* z = p ? z1 : z0;
  int t = threadIdx.x;
  *(float4*)&zs[t * 4] = *(const float4*)(z + (size_t)n * SEM + t * 4);
  __syncthreads();
  float acc = b1[t];
  for (int k = 0; k < SEM; ++k) acc += zs[k] * W1[k * 128 + t];
  red[t] = tanhf(acc) * W2[t];
  __syncthreads();
  for (int s = 64; s > 0; s >>= 1) {
    if (t < s) red[t] += red[t + s];
    __syncthreads();
  }
  if (t == 0) w[n * 2 + p] = red[0];
}

__global__ void combine(const float* __restrict__ z0, const float* __restrict__ z1,
                        const float* __restrict__ w, float* __restrict__ zf) {
  int i = blockIdx.x * 256 + threadIdx.x;
  if (i >= N_USER * SEM) return;
  int n = i >> 9;
  float w0 = w[n * 2], w1 = w[n * 2 + 1];
  float mx = fmaxf(w0, w1);
  float e0 = expf(w0 - mx), e1 = expf(w1 - mx);
  float inv = 1.f / (e0 + e1);
  zf[i] = (e0 * z0[i] + e1 * z1[i]) * inv;
}

extern "C" void kernel_launch(void* const* d_in, const int* in_sizes, int n_in,
                              void* d_out, int out_size, void* d_ws, size_t ws_size,
                              hipStream_t stream) {
  const float* x_user = (const float*)d_in[0];
  const float* x_llm  = (const float*)d_in[1];
  const int* src0 = (const int*)d_in[2];
  const int* dst0 = (const int*)d_in[3];
  const int* src1 = (const int*)d_in[4];
  const int* dst1 = (const int*)d_in[5];
  const float* Wsrc0 = (const float*)d_in[6];
  const float* Wdst0 = (const float*)d_in[7];
  const float* asrc0 = (const float*)d_in[8];
  const float* adst0 = (const float*)d_in[9];
  const float* bias0 = (const float*)d_in[10];
  const float* Wsrc1 = (const float*)d_in[11];
  const float* Wdst1 = (const float*)d_in[12];
  const float* asrc1 = (const float*)d_in[13];
  const float* adst1 = (const float*)d_in[14];
  const float* bias1 = (const float*)d_in[15];
  const float* semW1 = (const float*)d_in[16];
  const float* semb1 = (const float*)d_in[17];
  const float* semW2 = (const float*)d_in[18];
  const float* outW  = (const float*)d_in[19];
  const float* outb  = (const float*)d_in[20];

  float* ws = (float*)d_ws;
  size_t o = 0;
  float* hs0 = ws + o;  o += (size_t)N_LLM * SEM;
  float* hs1 = ws + o;  o += (size_t)N_USER * SEM;
  float* z0  = ws + o;  o += (size_t)N_USER * SEM;
  float* z1  = ws + o;  o += (size_t)N_USER * SEM;
  float* as0 = ws + o;  o += (size_t)N_LLM * 4;
  float* ad0 = ws + o;  o += (size_t)N_USER * 4;
  float* as1 = ws + o;  o += (size_t)N_USER * 4;
  float* ad1 = ws + o;  o += (size_t)N_USER * 4;
  unsigned* m0 = (unsigned*)(ws + o); o += (size_t)N_USER * 4;
  float* den0 = ws + o; o += (size_t)N_USER * 4;
  unsigned* m1 = (unsigned*)(ws + o); o += (size_t)N_USER * 4;
  float* den1 = ws + o; o += (size_t)N_USER * 4;
  float* ex0 = ws + o;  o += (size_t)NEDGE * 4;
  float* ex1 = ws + o;  o += (size_t)NEDGE * 4;
  float* vf0 = ws + o;  o += (size_t)IN_DIM * 4;
  float* vf1 = ws + o;  o += (size_t)IN_DIM * 4;
  float* wsc = ws + o;  o += (size_t)N_USER * 2;
  float* zfin = hs1;  // hs1 is dead after metapath-1 message pass

  // zero z0,z1,a*,m*,den* (contiguous span starting at z0)
  size_t zero_floats = (size_t)N_USER * SEM * 2 + (size_t)N_LLM * 4 + (size_t)N_USER * 4 * 7;
  hipMemsetAsync(z0, 0, zero_floats * sizeof(float), stream);

  // dense GEMMs (WMMA): block tile 64x128
  gemm_f32_wmma<<<dim3(SEM / 128, (N_LLM + 63) / 64), 256, 0, stream>>>(
      x_llm, Wsrc0, hs0, nullptr, N_LLM, SEM, IN_DIM);
  gemm_f32_wmma<<<dim3(SEM / 128, (N_USER + 63) / 64), 256, 0, stream>>>(
      x_user, Wsrc1, hs1, nullptr, N_USER, SEM, IN_DIM);

  // attention logit scalars (hd GEMMs algebraically folded away)
  rowdot_att<<<(N_LLM * 4 + 7) / 8, 256, 0, stream>>>(hs0, asrc0, as0, N_LLM);
  rowdot_att<<<(N_USER * 4 + 7) / 8, 256, 0, stream>>>(hs1, asrc1, as1, N_USER);
  rowdot_att<<<(IN_DIM * 4 + 7) / 8, 256, 0, stream>>>(Wdst0, adst0, vf0, IN_DIM);
  rowdot_att<<<(IN_DIM * 4 + 7) / 8, 256, 0, stream>>>(Wdst1, adst1, vf1, IN_DIM);
  matvec_fold<<<(N_USER * 4 + 7) / 8, 256, 0, stream>>>(x_user, vf0, ad0, N_USER);
  matvec_fold<<<(N_USER * 4 + 7) / 8, 256, 0, stream>>>(x_user, vf1, ad1, N_USER);

  // edge softmax + messages, metapath 0 then 1
  int eg = (NEDGE + 255) / 256;
  edge_max<<<eg, 256, 0, stream>>>(src0, dst0, as0, ad0, m0);
  edge_exp<<<eg, 256, 0, stream>>>(src0, dst0, as0, ad0, m0, ex0, den0);
  edge_msg<<<NEDGE / 2, 256, 0, stream>>>(src0, dst0, hs0, ex0, den0, z0);
  edge_max<<<eg, 256, 0, stream>>>(src1, dst1, as1, ad1, m1);
  edge_exp<<<eg, 256, 0, stream>>>(src1, dst1, as1, ad1, m1, ex1, den1);
  edge_msg<<<NEDGE / 2, 256, 0, stream>>>(src1, dst1, hs1, ex1, den1, z1);

  int ng = (N_USER * SEM + 255) / 256;
  bias_elu<<<ng, 256, 0, stream>>>(z0, bias0, N_USER * SEM);
  bias_elu<<<ng, 256, 0, stream>>>(z1, bias1, N_USER * SEM);

  // semantic attention + output projection (WMMA)
  sem_score<<<dim3(N_USER, 2), 128, 0, stream>>>(z0, z1, semW1, semb1, semW2, wsc);
  combine<<<ng, 256, 0, stream>>>(z0, z1, wsc, zfin);
  gemm_f32_wmma<<<dim3(OUT_DIM / 128, (N_USER + 63) / 64), 256, 0, stream>>>(
      zfin, outW, (float*)d_out, outb, N_USER, OUT_DIM, SEM);
}